// FeedForwardQuantum_65481071400845
// MI455X (gfx1250) — compile-verified
//
#include <hip/hip_runtime.h>
#include <stdint.h>

// out = cumprod(cos(cumprod(cos(x), -1)), -1), rows of 512 f32, 65536 rows.
// Memory-bound: 256 MiB traffic / 23.3 TB/s ~= 11 us floor. Strategy:
// wave32-per-row segmented scan, async global->LDS double-buffered streaming
// (CDNA5 ASYNCcnt path), native v_cos_f32, in-register second encode.

#define ROW_ELEMS     512
#define EPL           16      // elements per lane (512 / 32)
#define WAVES_PER_WG  8
#define WG_THREADS    256
#define LDS_ROW_BYTES 2560    // 32 lanes * (64B data + 16B pad) -> conflict-light ds_load_b128
#define LDS_CHUNK     640     // LDS bytes covered by one 512B global chunk (k)

typedef __attribute__((ext_vector_type(4))) float v4f;
typedef int v4i __attribute__((vector_size(16)));            // matches builtin param pointee
typedef __attribute__((address_space(3))) char   as3_char;
typedef __attribute__((address_space(3))) v4f    as3_v4f;
typedef __attribute__((address_space(3))) v4i    as3_v4i;
typedef __attribute__((address_space(1))) char   as1_char;
typedef __attribute__((address_space(1))) v4i    as1_v4i;

#if defined(__HIP_DEVICE_COMPILE__) && __has_builtin(__builtin_amdgcn_global_load_async_to_lds_b128)
#define HAVE_ASYNC_LDS 1
#else
#define HAVE_ASYNC_LDS 0
#endif

template <int N>
__device__ __forceinline__ void wait_asynccnt() {
#if __has_builtin(__builtin_amdgcn_s_wait_asynccnt)
  __builtin_amdgcn_s_wait_asynccnt((short)N);
#else
  asm volatile("s_wait_asynccnt %0" ::"i"(N) : "memory");
#endif
  asm volatile("" ::: "memory");  // keep ds reads below the wait
}

__device__ __forceinline__ void wait_dscnt0() {
#if __has_builtin(__builtin_amdgcn_s_wait_dscnt)
  __builtin_amdgcn_s_wait_dscnt(0);
#else
  asm volatile("s_wait_dscnt 0x0" ::: "memory");
#endif
  asm volatile("" ::: "memory");
}

// Two rounds of: per-element cos, lane-local cumprod(16), wave32 prefix
// product (5 shfl_up steps), scale. After round 1, h lives in registers in
// exactly the layout round 2 needs.
__device__ __forceinline__ void encode_twice(float v[EPL], int lane) {
#pragma unroll
  for (int pass = 0; pass < 2; ++pass) {
    float run = 1.0f;
#pragma unroll
    for (int i = 0; i < EPL; ++i) {
      run *= __cosf(v[i]);   // native v_cos_f32; args are N(0,1) / in [-1,1]
      v[i] = run;
    }
    float incl = run;        // inclusive scan of per-lane totals (wave32)
#pragma unroll
    for (int d = 1; d < 32; d <<= 1) {
      float u = __shfl_up(incl, d, 32);
      if (lane >= d) incl *= u;
    }
    float excl  = __shfl_up(incl, 1, 32);
    float scale = (lane == 0) ? 1.0f : excl;
#pragma unroll
    for (int i = 0; i < EPL; ++i) v[i] *= scale;
  }
}

__global__ __launch_bounds__(WG_THREADS)
void qff_scan_kernel(const float* __restrict__ x, float* __restrict__ out, int nRows) {
  const int lane   = threadIdx.x & 31;
  const int wv     = threadIdx.x >> 5;
  const int wave   = blockIdx.x * WAVES_PER_WG + wv;
  const int nWaves = gridDim.x * WAVES_PER_WG;

  float v[EPL];

#if HAVE_ASYNC_LDS
  __shared__ __align__(16) char lds[WAVES_PER_WG * 2 * LDS_ROW_BYTES];  // 40 KB/WG
  char* buf0 = &lds[(wv * 2 + 0) * LDS_ROW_BYTES];
  char* buf1 = &lds[(wv * 2 + 1) * LDS_ROW_BYTES];

  // Async DMA one 2KB row into this wave's LDS buffer: 4 x b128 (16B/lane).
  // Coalesced global chunk k, lane L -> padded LDS slot of owner lane
  // o = 8k + L/4 at byte 640k + 80*(L>>2) + 16*(L&3).
  auto issue = [&](int row, char* b) {
    as1_char* g = (as1_char*)(uintptr_t)(x + (size_t)row * ROW_ELEMS) + lane * 16;
    as3_char* l = (as3_char*)(uint32_t)(uintptr_t)b + (80u * (lane >> 2) + 16u * (lane & 3));
#pragma unroll
    for (int k = 0; k < 4; ++k)
      __builtin_amdgcn_global_load_async_to_lds_b128(
          (as1_v4i*)(g + k * 512), (as3_v4i*)(l + k * LDS_CHUNK), 0, 0);
  };
  // Lane L reads its contiguous 16 floats at byte 80*L (16B aligned, padded stride).
  auto load_lds = [&](char* b) {
    as3_v4f* p = (as3_v4f*)(uint32_t)((uintptr_t)b + 80u * lane);
#pragma unroll
    for (int i = 0; i < 4; ++i) {
      v4f t = p[i];
      v[4 * i + 0] = t.x; v[4 * i + 1] = t.y; v[4 * i + 2] = t.z; v[4 * i + 3] = t.w;
    }
  };
#endif

  auto load_global = [&](int row) {
    const v4f* g = (const v4f*)(x + (size_t)row * ROW_ELEMS) + lane * 4;
#pragma unroll
    for (int i = 0; i < 4; ++i) {
      v4f t = g[i];
      v[4 * i + 0] = t.x; v[4 * i + 1] = t.y; v[4 * i + 2] = t.z; v[4 * i + 3] = t.w;
    }
  };
  auto store_row = [&](int row) {
    v4f* o = (v4f*)(out + (size_t)row * ROW_ELEMS) + lane * 4;
#pragma unroll
    for (int i = 0; i < 4; ++i) {
      v4f t;
      t.x = v[4 * i + 0]; t.y = v[4 * i + 1]; t.z = v[4 * i + 2]; t.w = v[4 * i + 3];
      o[i] = t;
    }
  };

#if HAVE_ASYNC_LDS
  int row = wave;
  int pb  = 0;
  if (row < nRows) issue(row, buf0);
  for (; row < nRows; row += nWaves) {
    const int nxt = row + nWaves;
    char* cur = pb ? buf1 : buf0;
    char* oth = pb ? buf0 : buf1;
    if (nxt < nRows) {
      wait_dscnt0();            // WAR guard: prior ds reads of 'oth' complete
      issue(nxt, oth);          // overlap next row's DMA with this compute
      wait_asynccnt<4>();       // oldest 4 (== 'cur' row) done; 4 stay in flight
    } else {
      wait_asynccnt<0>();
    }
    load_lds(cur);
    encode_twice(v, lane);
    store_row(row);
    pb ^= 1;
  }
#else
  for (int row = wave; row < nRows; row += nWaves) {
    const int nxt = row + nWaves;
    if (nxt < nRows) {
      const char* pf = (const char*)(x + (size_t)nxt * ROW_ELEMS) + lane * 64;
      __builtin_prefetch(pf, 0, 1);   // global_prefetch_b8
    }
    load_global(row);
    encode_twice(v, lane);
    store_row(row);
  }
#endif
}

extern "C" void kernel_launch(void* const* d_in, const int* in_sizes, int n_in,
                              void* d_out, int out_size, void* d_ws, size_t ws_size,
                              hipStream_t stream) {
  (void)n_in; (void)out_size; (void)d_ws; (void)ws_size;
  const float* x = (const float*)d_in[0];
  float* out     = (float*)d_out;
  const int nRows = in_sizes[0] / ROW_ELEMS;   // 65536 for the reference shapes

  // ~8 rows per wave so the double-buffered async pipeline has depth.
  int waves  = (nRows + 7) / 8;
  int blocks = (waves + WAVES_PER_WG - 1) / WAVES_PER_WG;
  if (blocks < 1) blocks = 1;
  if (blocks > 8192) blocks = 8192;

  qff_scan_kernel<<<blocks, WG_THREADS, 0, stream>>>(x, out, nRows);
}